// PeakDetect_81801947119861
// MI455X (gfx1250) — compile-verified
//
#include <hip/hip_runtime.h>
#include <hip/hip_bf16.h>

#define AS1 __attribute__((address_space(1)))
#define AS3 __attribute__((address_space(3)))

typedef int v4i __attribute__((vector_size(16)));   // matches builtin's pointee type

#ifndef __has_builtin
#define __has_builtin(x) 0
#endif

#if __has_builtin(__builtin_amdgcn_global_load_async_to_lds_b128)
#define HAVE_ASYNC_LDS 1
#else
#define HAVE_ASYNC_LDS 0
#endif

__device__ __forceinline__ void async_fence_lds() {
#if __has_builtin(__builtin_amdgcn_s_wait_asynccnt)
    __builtin_amdgcn_s_wait_asynccnt(0);
#else
    asm volatile("s_wait_asynccnt 0" ::: "memory");
#endif
}

// Problem constants (fixed by setup_inputs)
#define BATCH 8
#define CH    16
#define H     512
#define W     512
#define MAXP  262144
#define TILE_ROWS 16
#define TILES_PER_IMG   (H / TILE_ROWS)          // 32
#define TILES_PER_BATCH (CH * TILES_PER_IMG)     // 512
#define NBLOCKS         (BATCH * TILES_PER_BATCH)// 4096
#define THREADS 256
#define PEAK_FLOATS ((size_t)BATCH * MAXP * 3)   // 6,291,456
#define PEAK_F4     (PEAK_FLOATS / 4)            // 1,572,864

// ---------------------------------------------------------------------------
// Stage an 18x512 halo tile (rows y0-1 .. y0+16) of one channel image into LDS.
// Out-of-range rows get -inf (matches reduce_window's -inf padding).
// Uses CDNA5 async global->LDS b128 copies when available (ASYNCcnt tracked).
// ---------------------------------------------------------------------------
__device__ __forceinline__ void stage_tile(const float* __restrict__ img,
                                           int y0, float* sm) {
    const int t = threadIdx.x;
    // 18 rows * 128 float4 per row = 2304 float4 transfers, 9 per thread.
    for (int i = t; i < (TILE_ROWS + 2) * (W / 4); i += THREADS) {
        const int r  = i >> 7;        // LDS row 0..17
        const int q  = i & 127;       // float4 index within row
        const int gy = y0 - 1 + r;    // global row
        float4* ldst = (float4*)(sm + r * W) + q;
        if (gy >= 0 && gy < H) {
            const float4* gsrc = (const float4*)(img + (size_t)gy * W) + q;
#if HAVE_ASYNC_LDS
            __builtin_amdgcn_global_load_async_to_lds_b128(
                (AS1 v4i*)gsrc, (AS3 v4i*)ldst, 0, 0);
#else
            *ldst = *gsrc;
#endif
        } else {
            *ldst = make_float4(-INFINITY, -INFINITY, -INFINITY, -INFINITY);
        }
    }
#if HAVE_ASYNC_LDS
    async_fence_lds();
#endif
    __syncthreads();
}

// ---------------------------------------------------------------------------
// Per-thread 32-element peak bitmap. Thread t covers flat tile offsets
// [t*32, t*32+32): row rt = t>>4 of the tile, columns (t&15)*32 .. +32.
// (thread-major order == flat order -> stable compaction via block scan)
// ---------------------------------------------------------------------------
__device__ __forceinline__ unsigned peak_mask32(const float* sm, int t) {
    const int rt = t >> 4;
    const int cb = (t & 15) << 5;
    const float* rm1 = sm + rt * W;        // row above
    const float* r0  = sm + (rt + 1) * W;  // center row
    const float* rp1 = sm + (rt + 2) * W;  // row below
    const float NI = -INFINITY;
    float um = (cb > 0) ? rm1[cb - 1] : NI;
    float cm = (cb > 0) ? r0 [cb - 1] : NI;
    float dm = (cb > 0) ? rp1[cb - 1] : NI;
    float u0 = rm1[cb], c0 = r0[cb], d0 = rp1[cb];
    unsigned m = 0;
    #pragma unroll
    for (int j = 0; j < 32; ++j) {
        const int xp = cb + j + 1;
        const float up = (xp < W) ? rm1[xp] : NI;
        const float cp = (xp < W) ? r0 [xp] : NI;
        const float dp = (xp < W) ? rp1[xp] : NI;
        const float v = c0;
        const bool peak = (v > 0.0f) &
                          (v >= um) & (v >= u0) & (v >= up) &
                          (v >= cm) &             (v >= cp) &
                          (v >= dm) & (v >= d0) & (v >= dp);
        m |= (peak ? 1u : 0u) << j;
        um = u0; u0 = up; cm = c0; c0 = cp; dm = d0; d0 = dp;
    }
    return m;
}

__device__ __forceinline__ void decode_block(int blk, int& b, int& ch, int& y0,
                                             const float* hm, const float*& img) {
    b = blk / TILES_PER_BATCH;
    const int tt = blk % TILES_PER_BATCH;
    ch = tt / TILES_PER_IMG;
    const int ty = tt % TILES_PER_IMG;
    y0 = ty * TILE_ROWS;
    img = hm + ((size_t)b * CH + ch) * ((size_t)H * W);
}

// ---------------------------------------------------------------------------
// Pass 1: per-tile peak counts.
// ---------------------------------------------------------------------------
__global__ void __launch_bounds__(THREADS)
count_kernel(const float* __restrict__ hm, unsigned* __restrict__ blockCounts) {
    __shared__ float sm[(TILE_ROWS + 2) * W];
    __shared__ unsigned red[THREADS];
    const int t = threadIdx.x;
    int b, ch, y0; const float* img;
    decode_block(blockIdx.x, b, ch, y0, hm, img);
    stage_tile(img, y0, sm);
    red[t] = __popc(peak_mask32(sm, t));
    __syncthreads();
    #pragma unroll
    for (int s = THREADS / 2; s > 0; s >>= 1) {
        if (t < s) red[t] += red[t + s];
        __syncthreads();
    }
    if (t == 0) blockCounts[blockIdx.x] = red[0];
}

// ---------------------------------------------------------------------------
// Pass 2: per-batch exclusive scan over 512 tile counts; emit counts[b].
// ---------------------------------------------------------------------------
__global__ void __launch_bounds__(TILES_PER_BATCH)
scan_kernel(const unsigned* __restrict__ blockCounts,
            unsigned* __restrict__ blockOffsets,
            float* __restrict__ out) {
    __shared__ unsigned s[TILES_PER_BATCH];
    const int b = blockIdx.x;
    const int t = threadIdx.x;
    const unsigned v = blockCounts[b * TILES_PER_BATCH + t];
    s[t] = v;
    __syncthreads();
    for (int off = 1; off < TILES_PER_BATCH; off <<= 1) {
        const unsigned add = (t >= off) ? s[t - off] : 0u;
        __syncthreads();
        s[t] += add;
        __syncthreads();
    }
    blockOffsets[b * TILES_PER_BATCH + t] = s[t] - v;   // exclusive prefix
    if (t == TILES_PER_BATCH - 1)
        out[PEAK_FLOATS + b] = (float)s[t];             // uncapped total count
}

// ---------------------------------------------------------------------------
// Pass 3: zero-fill the peaks region (padding semantics).
// ---------------------------------------------------------------------------
__global__ void zero_kernel(float4* __restrict__ p) {
    const size_t i = (size_t)blockIdx.x * blockDim.x + threadIdx.x;
    if (i < PEAK_F4) p[i] = make_float4(0.f, 0.f, 0.f, 0.f);
}

// ---------------------------------------------------------------------------
// Pass 4: stable scatter. Recompute masks (input hits the 192MB L2),
// block-local exclusive scan for stable ranks, write [x, y, conf].
// ---------------------------------------------------------------------------
__global__ void __launch_bounds__(THREADS)
write_kernel(const float* __restrict__ hm,
             const unsigned* __restrict__ blockOffsets,
             float* __restrict__ out) {
    __shared__ float sm[(TILE_ROWS + 2) * W];
    __shared__ unsigned s[THREADS];
    const int t = threadIdx.x;
    int b, ch, y0; const float* img;
    decode_block(blockIdx.x, b, ch, y0, hm, img);
    stage_tile(img, y0, sm);
    const unsigned mask = peak_mask32(sm, t);
    const unsigned cnt = __popc(mask);
    s[t] = cnt;
    __syncthreads();
    for (int off = 1; off < THREADS; off <<= 1) {
        const unsigned add = (t >= off) ? s[t - off] : 0u;
        __syncthreads();
        s[t] += add;
        __syncthreads();
    }
    unsigned base = blockOffsets[blockIdx.x] + (s[t] - cnt);
    if (mask == 0u) return;
    const int rt = t >> 4;
    const int cb = (t & 15) << 5;
    const float* r0 = sm + (rt + 1) * W;
    const int gy = y0 + rt;
    #pragma unroll 4
    for (int j = 0; j < 32; ++j) {
        if ((mask >> j) & 1u) {
            if (base < MAXP) {
                const size_t o = ((size_t)b * MAXP + base) * 3;
                out[o + 0] = (float)(cb + j);   // x = order % W
                out[o + 1] = (float)gy;         // y = (order // W) % H
                out[o + 2] = r0[cb + j];        // confidence
            }
            ++base;
        }
    }
}

extern "C" void kernel_launch(void* const* d_in, const int* in_sizes, int n_in,
                              void* d_out, int out_size, void* d_ws, size_t ws_size,
                              hipStream_t stream) {
    const float* hm = (const float*)d_in[0];
    float* out = (float*)d_out;
    unsigned* blockCounts  = (unsigned*)d_ws;
    unsigned* blockOffsets = blockCounts + NBLOCKS;

    zero_kernel<<<(unsigned)((PEAK_F4 + 255) / 256), 256, 0, stream>>>((float4*)out);
    count_kernel<<<NBLOCKS, THREADS, 0, stream>>>(hm, blockCounts);
    scan_kernel<<<BATCH, TILES_PER_BATCH, 0, stream>>>(blockCounts, blockOffsets, out);
    write_kernel<<<NBLOCKS, THREADS, 0, stream>>>(hm, blockOffsets, out);
}